// CopynetDecoderRNN_19628000543107
// MI455X (gfx1250) — compile-verified
//
#include <hip/hip_runtime.h>
#include <hip/hip_bf16.h>
#include <math.h>

#define B  256
#define L  512
#define E  200
#define H  256
#define DV 30000
#define V  50000
#define XDIM (2*H + E)   // 712

typedef float v2f __attribute__((ext_vector_type(2)));
typedef float v8f __attribute__((ext_vector_type(8)));

__device__ __forceinline__ float sigmoidf_(float x){ return 1.0f/(1.0f+__expf(-x)); }

// ---------------- K1: selective read -------------------------------------
__global__ void k_selective(const int* __restrict__ input_id,
                            const int* __restrict__ enc_ids,
                            const float* __restrict__ enc,
                            float* __restrict__ sel) {
  int b = blockIdx.x, t = threadIdx.x;
  __shared__ int match[L];
  __shared__ int cnt;
  if (t == 0) cnt = 0;
  __syncthreads();
  int id = input_id[b];
  int local = 0;
  for (int l = t; l < L; l += 256) { int m = (enc_ids[b*L+l] == id) ? 1 : 0; match[l] = m; local += m; }
  atomicAdd(&cnt, local);
  __syncthreads();
  float inv = (cnt > 0) ? 1.0f/(float)cnt : 0.0f;
  float acc = 0.f;
  for (int l = 0; l < L; ++l)
    if (match[l]) acc += enc[((size_t)b*L + l)*H + t];
  sel[b*H + t] = acc * inv;
}

// ---------------- K2: GRU cell (fused gx/gh/gates) ------------------------
__global__ void k_gru(const float* __restrict__ emb, const float* __restrict__ attn,
                      const float* __restrict__ sel, const float* __restrict__ hidden,
                      const float* __restrict__ W_ih, const float* __restrict__ W_hh,
                      const float* __restrict__ b_ih, const float* __restrict__ b_hh,
                      float* __restrict__ hws, float* __restrict__ h_out) {
  int b = blockIdx.x, t = threadIdx.x;
  __shared__ float x[XDIM];
  __shared__ float hp[H];
  for (int i = t; i < XDIM; i += 256) {
    float v;
    if (i < E)        v = emb[b*E + i];
    else if (i < E+H) v = attn[b*H + (i - E)];
    else              v = sel[b*H + (i - E - H)];
    x[i] = v;
  }
  hp[t] = hidden[b*H + t];
  __syncthreads();
  float gx[3], gh[3];
  #pragma unroll
  for (int g = 0; g < 3; ++g) {
    int j = g*H + t;
    const float* wi = W_ih + (size_t)j*XDIM;
    float a = b_ih[j];
    for (int k = 0; k < XDIM; ++k) a += x[k]*wi[k];
    gx[g] = a;
    const float* wh = W_hh + (size_t)j*H;
    float c = b_hh[j];
    for (int k = 0; k < H; ++k) c += hp[k]*wh[k];
    gh[g] = c;
  }
  float r = sigmoidf_(gx[0] + gh[0]);
  float z = sigmoidf_(gx[1] + gh[1]);
  float n = tanhf(gx[2] + r*gh[2]);
  float h = (1.0f - z)*n + z*hp[t];
  hws[b*H + t]   = h;
  h_out[b*H + t] = h;
}

// ---------------- K3: q = h @ W_attn.T + b --------------------------------
__global__ void k_q(const float* __restrict__ hws, const float* __restrict__ W_attn,
                    const float* __restrict__ b_attn, float* __restrict__ qws) {
  int b = blockIdx.x, t = threadIdx.x;
  __shared__ float hs[H];
  hs[t] = hws[b*H + t];
  __syncthreads();
  const float* w = W_attn + (size_t)t*H;
  float a = b_attn[t];
  for (int k = 0; k < H; ++k) a += hs[k]*w[k];
  qws[b*H + t] = a;
}

// ---------------- K4: attention scores + softmax + apply ------------------
__global__ void k_attn(const float* __restrict__ qws, const float* __restrict__ enc,
                       float* __restrict__ app) {
  int b = blockIdx.x, t = threadIdx.x;
  __shared__ float qs[H];
  __shared__ float sc[L];
  __shared__ float red[256];
  qs[t] = qws[b*H + t];
  __syncthreads();
  for (int l = t; l < L; l += 256) {
    const float* e = enc + ((size_t)b*L + l)*H;
    float a = 0.f;
    for (int k = 0; k < H; ++k) a += qs[k]*e[k];
    sc[l] = a;
  }
  __syncthreads();
  float m = -INFINITY;
  for (int l = t; l < L; l += 256) m = fmaxf(m, sc[l]);
  red[t] = m; __syncthreads();
  for (int s = 128; s > 0; s >>= 1) { if (t < s) red[t] = fmaxf(red[t], red[t+s]); __syncthreads(); }
  float mx = red[0]; __syncthreads();
  float sacc = 0.f;
  for (int l = t; l < L; l += 256) { float e = __expf(sc[l] - mx); sc[l] = e; sacc += e; }
  red[t] = sacc; __syncthreads();
  for (int s = 128; s > 0; s >>= 1) { if (t < s) red[t] += red[t+s]; __syncthreads(); }
  float inv = 1.0f/red[0];
  __syncthreads();
  float acc = 0.f;
  for (int l = 0; l < L; ++l) acc += sc[l]*enc[((size_t)b*L + l)*H + t];
  app[b*H + t] = acc*inv;
}

// ---------------- K5: cur_attention = tanh([app,h] @ W_comb.T + b) --------
__global__ void k_comb(const float* __restrict__ app, const float* __restrict__ hws,
                       const float* __restrict__ W_comb, const float* __restrict__ b_comb,
                       float* __restrict__ cat, float* __restrict__ ca_out) {
  int b = blockIdx.x, t = threadIdx.x;
  __shared__ float c[2*H];
  c[t]     = app[b*H + t];
  c[H + t] = hws[b*H + t];
  __syncthreads();
  const float* w = W_comb + (size_t)t*(2*H);
  float a = b_comb[t];
  for (int k = 0; k < 2*H; ++k) a += c[k]*w[k];
  float v = tanhf(a);
  cat[b*H + t]    = v;
  ca_out[b*H + t] = v;
}

// ---------------- K6: generate_score = cat @ W_gen.T + b_gen  (f32 WMMA) --
// one wave per 16x16 output tile; M over batch (16 tiles), N over DV (1875 tiles)
__global__ void k_gen_wmma(const float* __restrict__ cat, const float* __restrict__ W_gen,
                           const float* __restrict__ b_gen, float* __restrict__ gen) {
  int wave = (blockIdx.x*blockDim.x + threadIdx.x) >> 5;   // 30000 waves total
  int lane = threadIdx.x & 31;
  int mTile = wave & 15;          // B/16
  int nTile = wave >> 4;          // DV/16
  int lx = lane & 15;
  int khalf = (lane >> 4) * 2;    // 0 or 2
  const float* aPtr = cat   + ((size_t)(mTile*16 + lx))*H + khalf;
  const float* bPtr = W_gen + ((size_t)(nTile*16 + lx))*H + khalf;
  v8f acc = {};
  #pragma unroll 8
  for (int k = 0; k < H; k += 4) {
    v2f a  = *(const v2f*)(aPtr + k);
    v2f bb = *(const v2f*)(bPtr + k);
    acc = __builtin_amdgcn_wmma_f32_16x16x4_f32(false, a, false, bb, (short)0, acc, false, false);
  }
  int n = nTile*16 + lx;
  int mBase = mTile*16 + (lane >> 4)*8;
  float bias = b_gen[n];
  #pragma unroll
  for (int i = 0; i < 8; ++i)
    gen[(size_t)(mBase + i)*DV + n] = acc[i] + bias;
}

// ---------------- K7: fused copy score (f32 WMMA, no 134MB intermediate) --
// one wave per 16-row tile of (B*L); inner loop over the 16 N-tiles of H.
__global__ void k_copy_wmma(const float* __restrict__ enc, const float* __restrict__ W_copy,
                            const float* __restrict__ b_copy, const float* __restrict__ cat,
                            float* __restrict__ cps) {
  int wave = (blockIdx.x*blockDim.x + threadIdx.x) >> 5;   // 8192 row tiles
  int lane = threadIdx.x & 31;
  int lx = lane & 15;
  int khalf = (lane >> 4) * 2;
  int rowBase = wave*16;            // row in flattened (B*L); tiles never cross b (512%16==0)
  int b = rowBase >> 9;             // / L
  const float* aPtr = enc + ((size_t)rowBase + lx)*H + khalf;
  float sum8[8];
  #pragma unroll
  for (int i = 0; i < 8; ++i) sum8[i] = 0.f;
  for (int nt = 0; nt < H/16; ++nt) {
    int n = nt*16 + lx;
    const float* bPtr = W_copy + (size_t)n*H + khalf;
    v8f acc = {};
    #pragma unroll 8
    for (int k = 0; k < H; k += 4) {
      v2f a  = *(const v2f*)(aPtr + k);
      v2f bb = *(const v2f*)(bPtr + k);
      acc = __builtin_amdgcn_wmma_f32_16x16x4_f32(false, a, false, bb, (short)0, acc, false, false);
    }
    float ca = cat[b*H + n];
    float bc = b_copy[n];
    #pragma unroll
    for (int i = 0; i < 8; ++i) sum8[i] += ca * sigmoidf_(acc[i] + bc);
  }
  // reduce across the 16 N-lanes of each half-wave
  #pragma unroll
  for (int i = 0; i < 8; ++i) {
    float v = sum8[i];
    v += __shfl_xor(v, 1, 32);
    v += __shfl_xor(v, 2, 32);
    v += __shfl_xor(v, 4, 32);
    v += __shfl_xor(v, 8, 32);
    sum8[i] = v;
  }
  if (lx == 0) {
    int mOff = (lane >> 4)*8;
    for (int i = 0; i < 8; ++i) cps[rowBase + mOff + i] = sum8[i];
  }
}

// ---------------- K8a: softmax stats over concat(gen[30000], copy[512]) ---
__global__ void k_stats(const float* __restrict__ gen, const float* __restrict__ cps,
                        float* __restrict__ rmax, float* __restrict__ rsum) {
  int b = blockIdx.x, t = threadIdx.x;
  __shared__ float red[256];
  float m = -INFINITY;
  for (int i = t; i < DV + L; i += 256) {
    float v = (i < DV) ? gen[(size_t)b*DV + i] : cps[b*L + (i - DV)];
    m = fmaxf(m, v);
  }
  red[t] = m; __syncthreads();
  for (int s = 128; s > 0; s >>= 1) { if (t < s) red[t] = fmaxf(red[t], red[t+s]); __syncthreads(); }
  float mx = red[0]; __syncthreads();
  float sacc = 0.f;
  for (int i = t; i < DV + L; i += 256) {
    float v = (i < DV) ? gen[(size_t)b*DV + i] : cps[b*L + (i - DV)];
    sacc += __expf(v - mx);
  }
  red[t] = sacc; __syncthreads();
  for (int s = 128; s > 0; s >>= 1) { if (t < s) red[t] += red[t+s]; __syncthreads(); }
  if (t == 0) { rmax[b] = mx; rsum[b] = red[0]; }
}

// ---------------- K8b: write generate probs, zero copy-only region --------
__global__ void k_writegen(const float* __restrict__ gen, const float* __restrict__ rmax,
                           const float* __restrict__ rsum, float* __restrict__ out) {
  size_t idx = (size_t)blockIdx.x*blockDim.x + threadIdx.x;
  if (idx >= (size_t)B*V) return;
  int b = (int)(idx / V);
  int v = (int)(idx % V);
  float val = 0.f;
  if (v < DV) val = __expf(gen[(size_t)b*DV + v] - rmax[b]) / rsum[b];
  out[idx] = val;
}

// ---------------- K8c: scatter-add copy probs -----------------------------
__global__ void k_scatter(const float* __restrict__ cps, const int* __restrict__ enc_ids,
                          const float* __restrict__ rmax, const float* __restrict__ rsum,
                          float* __restrict__ out) {
  int idx = blockIdx.x*blockDim.x + threadIdx.x;    // B*L
  int b = idx / L;
  float p = __expf(cps[idx] - rmax[b]) / rsum[b];
  atomicAdd(&out[(size_t)b*V + enc_ids[idx]], p);
}

// ---------------- K8d: log / -inf -----------------------------------------
__global__ void k_log(float* __restrict__ out) {
  size_t idx = (size_t)blockIdx.x*blockDim.x + threadIdx.x;
  if (idx >= (size_t)B*V) return;
  float v = out[idx];
  out[idx] = (v > 0.f) ? __logf(v) : -INFINITY;
}

extern "C" void kernel_launch(void* const* d_in, const int* in_sizes, int n_in,
                              void* d_out, int out_size, void* d_ws, size_t ws_size,
                              hipStream_t stream) {
  (void)in_sizes; (void)n_in; (void)out_size; (void)ws_size;
  const int*   input_id = (const int*)  d_in[0];
  const float* emb      = (const float*)d_in[1];
  const float* enc      = (const float*)d_in[2];
  const int*   enc_ids  = (const int*)  d_in[3];
  const float* hidden   = (const float*)d_in[4];
  const float* attn     = (const float*)d_in[5];
  const float* W_ih     = (const float*)d_in[6];
  const float* W_hh     = (const float*)d_in[7];
  const float* b_ih     = (const float*)d_in[8];
  const float* b_hh     = (const float*)d_in[9];
  const float* W_attn   = (const float*)d_in[10];
  const float* b_attn   = (const float*)d_in[11];
  const float* W_comb   = (const float*)d_in[12];
  const float* b_comb   = (const float*)d_in[13];
  const float* W_gen    = (const float*)d_in[14];
  const float* b_gen    = (const float*)d_in[15];
  const float* W_copy   = (const float*)d_in[16];
  const float* b_copy   = (const float*)d_in[17];

  float* out    = (float*)d_out;                       // (B, V)
  float* h_out  = out + (size_t)B*V;                   // (B, 1, H)
  float* ca_out = h_out + (size_t)B*H;                 // (B, 1, H)

  float* ws   = (float*)d_ws;
  float* sel  = ws;                          // B*H
  float* hws  = sel  + (size_t)B*H;          // B*H
  float* qws  = hws  + (size_t)B*H;          // B*H
  float* app  = qws  + (size_t)B*H;          // B*H
  float* cat  = app  + (size_t)B*H;          // B*H  (cur_attention)
  float* gen  = cat  + (size_t)B*H;          // B*DV
  float* cps  = gen  + (size_t)B*DV;         // B*L
  float* rmax = cps  + (size_t)B*L;          // B
  float* rsum = rmax + B;                    // B

  k_selective<<<B, 256, 0, stream>>>(input_id, enc_ids, enc, sel);
  k_gru      <<<B, 256, 0, stream>>>(emb, attn, sel, hidden, W_ih, W_hh, b_ih, b_hh, hws, h_out);
  k_q        <<<B, 256, 0, stream>>>(hws, W_attn, b_attn, qws);
  k_attn     <<<B, 256, 0, stream>>>(qws, enc, app);
  k_comb     <<<B, 256, 0, stream>>>(app, hws, W_comb, b_comb, cat, ca_out);

  k_gen_wmma <<<(16*(DV/16))/8, 256, 0, stream>>>(cat, W_gen, b_gen, gen);   // 3750 blocks
  k_copy_wmma<<<((B*L)/16)/8,   256, 0, stream>>>(enc, W_copy, b_copy, cat, cps); // 1024 blocks

  k_stats    <<<B, 256, 0, stream>>>(gen, cps, rmax, rsum);
  k_writegen <<<(B*V)/256, 256, 0, stream>>>(gen, rmax, rsum, out);   // 50000 blocks
  k_scatter  <<<(B*L)/256, 256, 0, stream>>>(cps, enc_ids, rmax, rsum, out);
  k_log      <<<(B*V)/256, 256, 0, stream>>>(out);
}